// SAGELayer_3985729651443
// MI455X (gfx1250) — compile-verified
//
#include <hip/hip_runtime.h>

typedef __attribute__((ext_vector_type(2))) float v2f;
typedef __attribute__((ext_vector_type(4))) float v4f;
typedef __attribute__((ext_vector_type(8))) float v8f;

#define D 64        // D_IN == D_EDGE == D_OUT
#define K 128       // concat feature dim
#define KSTEPS 32   // K / 4 (one V_WMMA_F32_16X16X4_F32 per step per N-tile)
#define WAVES_PER_BLOCK 4
#define EDGE_GRID 2048   // persistent blocks for the edge kernel

// ---------------------------------------------------------------------------
// Kernel 1: per-edge message GEMM  m[e] = W_msg * concat(nfeats[src[e]], efeats[e]) + b
// then atomic scatter-add into sums[dst[e]] and degree count.
// One wave32 handles a tile of 16 edges: C(16x64) = A(16x128) x B(128x64).
// Grid-stride persistent so W_msg is staged to LDS once per block, not per tile.
// ---------------------------------------------------------------------------
__global__ __launch_bounds__(128) void sage_edge_msg(
    const float* __restrict__ nfeats, const float* __restrict__ efeats,
    const int* __restrict__ src, const int* __restrict__ dst,
    const float* __restrict__ W_msg, const float* __restrict__ b_msg,
    float* __restrict__ sums, float* __restrict__ deg, int n_edges)
{
    __shared__ float lW[D * K];                      // 32 KB: W_msg row-major [64][128]
    {
        const v4f* wsrc = (const v4f*)W_msg;
        v4f*       wdst = (v4f*)lW;
        for (int i = threadIdx.x; i < (D * K) / 4; i += blockDim.x)
            wdst[i] = wsrc[i];
    }
    __syncthreads();

    const int wave   = threadIdx.x >> 5;
    const int lane   = threadIdx.x & 31;
    const int row    = lane & 15;                    // A row / B column-in-tile
    const int dlt    = (lane < 16) ? 0 : 2;          // K offset inside each 4-chunk (ISA A layout)
    const int mbase  = (lane < 16) ? 0 : 8;          // C/D row base for this lane half
    const int etiles = (n_edges + 15) / 16;
    const int stride = gridDim.x * WAVES_PER_BLOCK;

    // per-lane bias values for the 4 N-tiles (uniform across tiles)
    const float bias0 = b_msg[0 * 16 + row];
    const float bias1 = b_msg[1 * 16 + row];
    const float bias2 = b_msg[2 * 16 + row];
    const float bias3 = b_msg[3 * 16 + row];

    for (int tile = blockIdx.x * WAVES_PER_BLOCK + wave; tile < etiles; tile += stride) {
        const int e0   = tile * 16;
        const int e_ld = min(e0 + row, n_edges - 1); // clamp: garbage rows never stored
        const int s    = src[e_ld];

        // prefetch the next tile's streamed data while we compute this one
        {
            const int en = min(e0 + stride * 16 + row, n_edges - 1);
            __builtin_prefetch(efeats + (size_t)en * D, 0, 1);
            __builtin_prefetch(src + en, 0, 1);
            __builtin_prefetch(dst + en, 0, 1);
        }

        const float* baseN = nfeats + (size_t)s    * D + dlt;  // k in [0,64): gathered node feats
        const float* baseE = efeats + (size_t)e_ld * D + dlt;  // k in [64,128): edge feats

        v8f acc0 = {}, acc1 = {}, acc2 = {}, acc3 = {};

        #pragma unroll 4
        for (int kk = 0; kk < KSTEPS; ++kk) {
            v2f a = (kk < 16) ? *(const v2f*)(baseN + kk * 4)
                              : *(const v2f*)(baseE + (kk - 16) * 4);
            const int kb = kk * 4 + dlt;
            v2f b0 = *(const v2f*)(lW + (0 * 16 + row) * K + kb);
            v2f b1 = *(const v2f*)(lW + (1 * 16 + row) * K + kb);
            v2f b2 = *(const v2f*)(lW + (2 * 16 + row) * K + kb);
            v2f b3 = *(const v2f*)(lW + (3 * 16 + row) * K + kb);
            acc0 = __builtin_amdgcn_wmma_f32_16x16x4_f32(false, a, false, b0, (short)0, acc0, false, false);
            acc1 = __builtin_amdgcn_wmma_f32_16x16x4_f32(false, a, false, b1, (short)0, acc1, false, false);
            acc2 = __builtin_amdgcn_wmma_f32_16x16x4_f32(false, a, false, b2, (short)0, acc2, false, false);
            acc3 = __builtin_amdgcn_wmma_f32_16x16x4_f32(false, a, false, b3, (short)0, acc3, false, false);
        }

        // C/D layout: VGPR v, lanes 0-15 -> row v, lanes 16-31 -> row v+8; col = nt*16 + row
        #pragma unroll
        for (int v = 0; v < 8; ++v) {
            const int ee = e0 + mbase + v;
            if (ee < n_edges) {
                const int d = dst[ee];
                float* base = sums + (size_t)d * D + row;
                __hip_atomic_fetch_add(base + 0 * 16, acc0[v] + bias0,
                                       __ATOMIC_RELAXED, __HIP_MEMORY_SCOPE_AGENT);
                __hip_atomic_fetch_add(base + 1 * 16, acc1[v] + bias1,
                                       __ATOMIC_RELAXED, __HIP_MEMORY_SCOPE_AGENT);
                __hip_atomic_fetch_add(base + 2 * 16, acc2[v] + bias2,
                                       __ATOMIC_RELAXED, __HIP_MEMORY_SCOPE_AGENT);
                __hip_atomic_fetch_add(base + 3 * 16, acc3[v] + bias3,
                                       __ATOMIC_RELAXED, __HIP_MEMORY_SCOPE_AGENT);
            }
        }

        if (lane < 16 && (e0 + lane) < n_edges)
            __hip_atomic_fetch_add(&deg[dst[e0 + lane]], 1.0f,
                                   __ATOMIC_RELAXED, __HIP_MEMORY_SCOPE_AGENT);
    }
}

// ---------------------------------------------------------------------------
// Kernel 2: per-node apply  h = relu(W_apply * concat(nfeats[i], sums[i]/deg[i]) + b)
// ---------------------------------------------------------------------------
__global__ __launch_bounds__(128) void sage_apply(
    const float* __restrict__ nfeats, const float* __restrict__ sums,
    const float* __restrict__ deg, const float* __restrict__ W_apply,
    const float* __restrict__ b_apply, float* __restrict__ out, int n_nodes)
{
    __shared__ float lW[D * K];                      // 32 KB: W_apply [64][128]
    {
        const v4f* wsrc = (const v4f*)W_apply;
        v4f*       wdst = (v4f*)lW;
        for (int i = threadIdx.x; i < (D * K) / 4; i += blockDim.x)
            wdst[i] = wsrc[i];
    }
    __syncthreads();

    const int wave = threadIdx.x >> 5;
    const int lane = threadIdx.x & 31;
    const int tile = blockIdx.x * WAVES_PER_BLOCK + wave;
    const int i0   = tile * 16;
    if (i0 >= n_nodes) return;                       // wave-uniform: EXEC all-ones at WMMA

    const int row = lane & 15;
    const int dlt = (lane < 16) ? 0 : 2;
    const int i_ld = min(i0 + row, n_nodes - 1);

    const float dg  = deg[i_ld];
    const float inv = (dg > 0.0f) ? (1.0f / dg) : 0.0f;   // isolated nodes -> h_neigh = 0

    const float* baseN = nfeats + (size_t)i_ld * D + dlt; // k in [0,64): self feats
    const float* baseS = sums   + (size_t)i_ld * D + dlt; // k in [64,128): mean aggregate

    const float bias0 = b_apply[0 * 16 + row];
    const float bias1 = b_apply[1 * 16 + row];
    const float bias2 = b_apply[2 * 16 + row];
    const float bias3 = b_apply[3 * 16 + row];

    v8f acc0 = {}, acc1 = {}, acc2 = {}, acc3 = {};

    #pragma unroll 4
    for (int kk = 0; kk < KSTEPS; ++kk) {
        v2f a;
        if (kk < 16) {
            a = *(const v2f*)(baseN + kk * 4);
        } else {
            v2f s2 = *(const v2f*)(baseS + (kk - 16) * 4);
            a = s2 * inv;
        }
        const int kb = kk * 4 + dlt;
        v2f b0 = *(const v2f*)(lW + (0 * 16 + row) * K + kb);
        v2f b1 = *(const v2f*)(lW + (1 * 16 + row) * K + kb);
        v2f b2 = *(const v2f*)(lW + (2 * 16 + row) * K + kb);
        v2f b3 = *(const v2f*)(lW + (3 * 16 + row) * K + kb);
        acc0 = __builtin_amdgcn_wmma_f32_16x16x4_f32(false, a, false, b0, (short)0, acc0, false, false);
        acc1 = __builtin_amdgcn_wmma_f32_16x16x4_f32(false, a, false, b1, (short)0, acc1, false, false);
        acc2 = __builtin_amdgcn_wmma_f32_16x16x4_f32(false, a, false, b2, (short)0, acc2, false, false);
        acc3 = __builtin_amdgcn_wmma_f32_16x16x4_f32(false, a, false, b3, (short)0, acc3, false, false);
    }

    const int mbase = (lane < 16) ? 0 : 8;
    #pragma unroll
    for (int v = 0; v < 8; ++v) {
        const int ii = i0 + mbase + v;
        if (ii < n_nodes) {
            float* o = out + (size_t)ii * D + row;
            o[0 * 16] = fmaxf(acc0[v] + bias0, 0.0f);
            o[1 * 16] = fmaxf(acc1[v] + bias1, 0.0f);
            o[2 * 16] = fmaxf(acc2[v] + bias2, 0.0f);
            o[3 * 16] = fmaxf(acc3[v] + bias3, 0.0f);
        }
    }
}

// ---------------------------------------------------------------------------
extern "C" void kernel_launch(void* const* d_in, const int* in_sizes, int n_in,
                              void* d_out, int out_size, void* d_ws, size_t ws_size,
                              hipStream_t stream) {
    const float* nfeats  = (const float*)d_in[0];
    const float* efeats  = (const float*)d_in[1];
    const int*   src     = (const int*)  d_in[2];
    const int*   dst     = (const int*)  d_in[3];
    const float* W_msg   = (const float*)d_in[4];
    const float* b_msg   = (const float*)d_in[5];
    const float* W_apply = (const float*)d_in[6];
    const float* b_apply = (const float*)d_in[7];

    const int n_nodes = in_sizes[0] / D;   // nfeats: [N,1,64]
    const int n_edges = in_sizes[2];       // src: [E]

    float* sums = (float*)d_ws;                       // [N,64] accumulators
    float* deg  = sums + (size_t)n_nodes * D;         // [N] degree counts
    size_t zbytes = ((size_t)n_nodes * D + (size_t)n_nodes) * sizeof(float);
    if (zbytes > ws_size) zbytes = ws_size;
    hipMemsetAsync(d_ws, 0, zbytes, stream);          // graph-capture safe

    const int etiles  = (n_edges + 15) / 16;
    int eblocks = (etiles + WAVES_PER_BLOCK - 1) / WAVES_PER_BLOCK;
    if (eblocks > EDGE_GRID) eblocks = EDGE_GRID;     // persistent grid-stride
    sage_edge_msg<<<eblocks, 32 * WAVES_PER_BLOCK, 0, stream>>>(
        nfeats, efeats, src, dst, W_msg, b_msg, sums, deg, n_edges);

    const int ntiles  = (n_nodes + 15) / 16;
    const int nblocks = (ntiles + WAVES_PER_BLOCK - 1) / WAVES_PER_BLOCK;
    sage_apply<<<nblocks, 32 * WAVES_PER_BLOCK, 0, stream>>>(
        nfeats, sums, deg, W_apply, b_apply, (float*)d_out, n_nodes);
}